// TwoStagePKRouter_9620726743480
// MI455X (gfx1250) — compile-verified
//
#include <hip/hip_runtime.h>
#include <hip/hip_bf16.h>
#include <math.h>

#define N_TOK   16384
#define DIM     2048
#define SQK     64
#define NEXP    4096
#define TOPK    8
#define GR      16
#define SCOLS   144   // 64 (s1) + 64 (s2) + 16 (q_g)

typedef float v2f __attribute__((ext_vector_type(2)));
typedef float v4f __attribute__((ext_vector_type(4)));
typedef float v8f __attribute__((ext_vector_type(8)));

// ---------------------------------------------------------------------------
// Kernel 1: S[n,0:64] = x@W1.T ; S[n,64:128] = x@W2.T ; S[n,128:144] = x@Wg.T
// One wave owns one 16-row tile of x and ALL 9 sixteen-column blocks of S
// (9 x v8f accumulators). The A fragment is loaded once per K-step and fed
// to 9 V_WMMA_F32_16X16X4_F32 ops, so x is read from HBM exactly once.
// Layouts (ISA 7.12.2, f32):
//   A 16x4 : a.x = X[M=lane%16][k0+2*(lane/16)], a.y = +1  (contiguous pair)
//   B 4x16 : B[k][n] = W[n][k] -> same contiguous float2 from row-major W
//   C 16x16: acc[r] -> S[rowBase + r + 8*(lane/16)][cb*16 + lane%16]
// ---------------------------------------------------------------------------
__global__ __launch_bounds__(256)
void pk_gemm_scores(const float* __restrict__ x,
                    const float* __restrict__ W1,
                    const float* __restrict__ W2,
                    const float* __restrict__ Wg,
                    float* __restrict__ S)
{
    const int tid     = threadIdx.x;
    const int lane    = tid & 31;
    const int lo      = lane & 15;
    const int hi      = lane >> 4;
    const int waveId  = blockIdx.x * 8 + (tid >> 5);   // 1024 waves, 1 per row tile
    const int rowBase = waveId * 16;

    const size_t rowA = (size_t)(rowBase + lo) * DIM + 2 * hi;  // x fragment base
    const size_t rowW = (size_t)lo * DIM + 2 * hi;              // weight fragment base

    v8f acc[9] = {};

    #pragma unroll 2
    for (int k0 = 0; k0 < DIM; k0 += 4) {
        const v2f a = *(const v2f*)(x + rowA + k0);

        #pragma unroll
        for (int cb = 0; cb < 4; ++cb) {                 // W1 blocks: cols 0..63
            v2f b = *(const v2f*)(W1 + rowW + (size_t)cb * 16 * DIM + k0);
            acc[cb] = __builtin_amdgcn_wmma_f32_16x16x4_f32(
                false, a, false, b, (short)0, acc[cb], false, false);
        }
        #pragma unroll
        for (int cb = 0; cb < 4; ++cb) {                 // W2 blocks: cols 64..127
            v2f b = *(const v2f*)(W2 + rowW + (size_t)cb * 16 * DIM + k0);
            acc[4 + cb] = __builtin_amdgcn_wmma_f32_16x16x4_f32(
                false, a, false, b, (short)0, acc[4 + cb], false, false);
        }
        {                                                // Wg block: cols 128..143
            v2f b = *(const v2f*)(Wg + rowW + k0);
            acc[8] = __builtin_amdgcn_wmma_f32_16x16x4_f32(
                false, a, false, b, (short)0, acc[8], false, false);
        }
    }

    #pragma unroll
    for (int cb = 0; cb < 9; ++cb) {
        #pragma unroll
        for (int r = 0; r < 8; ++r) {
            S[(size_t)(rowBase + r + 8 * hi) * SCOLS + cb * 16 + lo] = acc[cb][r];
        }
    }
}

// ---------------------------------------------------------------------------
// Kernel 2: select_scores[n, i*64+j] = s1[n,i] + s2[n,j]   (256 MB stream-out)
// One block per row; s1/s2 staged in LDS; float4 non-temporal stores.
// ---------------------------------------------------------------------------
__global__ __launch_bounds__(256)
void pk_expand_scores(const float* __restrict__ S, float* __restrict__ scores)
{
    __shared__ float s1s[SQK];
    __shared__ float s2s[SQK];
    const int n = blockIdx.x;
    const int t = threadIdx.x;
    if (t < 128) {
        float v = S[(size_t)n * SCOLS + t];
        if (t < SQK) s1s[t] = v; else s2s[t - SQK] = v;
    }
    __syncthreads();

    const float a  = s1s[t >> 2];          // i = (t*16)/64 constant per thread
    const int   j0 = (t & 3) * 16;
    float* out = scores + (size_t)n * NEXP + t * 16;
    #pragma unroll
    for (int u = 0; u < 16; u += 4) {
        v4f v = { a + s2s[j0 + u],     a + s2s[j0 + u + 1],
                  a + s2s[j0 + u + 2], a + s2s[j0 + u + 3] };
        __builtin_nontemporal_store(v, (v4f*)(out + u));
    }
}

// ---------------------------------------------------------------------------
// Kernel 3: per-row product-key top-8, low-rank gate at winners, softmax.
// Global top-8 of the outer-sum grid must come from top8(s1) x top8(s2).
// Tie-breaking matches jax.lax.top_k: value desc, index asc.
// ---------------------------------------------------------------------------
__global__ __launch_bounds__(128)
void pk_router(const float* __restrict__ S, const float* __restrict__ G,
               float* __restrict__ out_idx, float* __restrict__ out_w)
{
    const int n = blockIdx.x * 128 + threadIdx.x;
    if (n >= N_TOK) return;
    const float* srow = S + (size_t)n * SCOLS;

    float v1[TOPK]; int i1[TOPK];
    float v2[TOPK]; int i2[TOPK];
    #pragma unroll
    for (int t = 0; t < TOPK; ++t) {
        v1[t] = -INFINITY; i1[t] = 0;
        v2[t] = -INFINITY; i2[t] = 0;
    }

    // streaming top-8 of s1 and s2 (strict > keeps lower index first on ties)
    for (int i = 0; i < SQK; ++i) {
        float v = srow[i];
        if (v > v1[TOPK - 1]) {
            int p = TOPK - 1;
            while (p > 0 && v > v1[p - 1]) { v1[p] = v1[p - 1]; i1[p] = i1[p - 1]; --p; }
            v1[p] = v; i1[p] = i;
        }
        float w = srow[SQK + i];
        if (w > v2[TOPK - 1]) {
            int p = TOPK - 1;
            while (p > 0 && w > v2[p - 1]) { v2[p] = v2[p - 1]; i2[p] = i2[p - 1]; --p; }
            v2[p] = w; i2[p] = i;
        }
    }

    // top-8 over the 64 candidate sums, tie-break by smaller flat index
    float cv[TOPK]; int ce[TOPK];
    #pragma unroll
    for (int t = 0; t < TOPK; ++t) { cv[t] = -INFINITY; ce[t] = 0x7fffffff; }
    for (int a = 0; a < TOPK; ++a) {
        for (int b = 0; b < TOPK; ++b) {
            float v = v1[a] + v2[b];
            int   e = i1[a] * SQK + i2[b];
            if ((v > cv[TOPK - 1]) || (v == cv[TOPK - 1] && e < ce[TOPK - 1])) {
                int p = TOPK - 1;
                while (p > 0 && ((v > cv[p - 1]) || (v == cv[p - 1] && e < ce[p - 1]))) {
                    cv[p] = cv[p - 1]; ce[p] = ce[p - 1]; --p;
                }
                cv[p] = v; ce[p] = e;
            }
        }
    }

    // low-rank gate only at the 8 winners: g = q_g . G[e]
    float q[GR];
    #pragma unroll
    for (int r = 0; r < GR; ++r) q[r] = srow[128 + r];

    float comb[TOPK];
    float m = -INFINITY;
    #pragma unroll
    for (int t = 0; t < TOPK; ++t) {
        const float* g = G + (size_t)ce[t] * GR;
        float acc = 0.f;
        #pragma unroll
        for (int r = 0; r < GR; ++r) acc = fmaf(q[r], g[r], acc);
        comb[t] = cv[t] + acc;
        m = fmaxf(m, comb[t]);
    }
    float ssum = 0.f;
    #pragma unroll
    for (int t = 0; t < TOPK; ++t) { comb[t] = __expf(comb[t] - m); ssum += comb[t]; }
    const float inv = 1.f / ssum;
    #pragma unroll
    for (int t = 0; t < TOPK; ++t) {
        out_idx[(size_t)n * TOPK + t] = (float)ce[t];
        out_w  [(size_t)n * TOPK + t] = comb[t] * inv;
    }
}

// ---------------------------------------------------------------------------
extern "C" void kernel_launch(void* const* d_in, const int* in_sizes, int n_in,
                              void* d_out, int out_size, void* d_ws, size_t ws_size,
                              hipStream_t stream)
{
    const float* x  = (const float*)d_in[0];
    const float* W1 = (const float*)d_in[1];
    const float* W2 = (const float*)d_in[2];
    const float* Wg = (const float*)d_in[3];
    const float* G  = (const float*)d_in[4];

    float* out_idx    = (float*)d_out;                       // [N, 8] indices (as float)
    float* out_w      = out_idx + (size_t)N_TOK * TOPK;      // [N, 8] weights
    float* out_scores = out_w   + (size_t)N_TOK * TOPK;      // [N, 4096] select_scores

    float* S = (float*)d_ws;                                 // [N, 144] fused s1|s2|q_g

    // 1024 waves, one per 16-row tile; 8 waves per 256-thread block
    pk_gemm_scores<<<dim3(N_TOK / 16 / 8), 256, 0, stream>>>(x, W1, W2, Wg, S);
    pk_expand_scores<<<dim3(N_TOK), 256, 0, stream>>>(S, out_scores);
    pk_router<<<dim3(N_TOK / 128), 128, 0, stream>>>(S, G, out_idx, out_w);
}